// GraphUpdate_87935160418348
// MI455X (gfx1250) — compile-verified
//
#include <hip/hip_runtime.h>
#include <hip/hip_bf16.h>

// ---------------- problem constants ----------------
#define NN     10000
#define EE     320000
#define NC_    9
#define NODE_C_ 35
#define HEADS_ 8

typedef __attribute__((ext_vector_type(2))) float v2f;
typedef __attribute__((ext_vector_type(8))) float v8f;
typedef int vint4 __attribute__((vector_size(16)));   // matches builtin's pointee type

#define AS1 __attribute__((address_space(1)))
#define AS3 __attribute__((address_space(3)))

#if __has_builtin(__builtin_amdgcn_global_load_async_to_lds_b128)
#define HAVE_ASYNC_LDS 1
#endif

// 16-byte global -> LDS copy. Async (ASYNCcnt-tracked, no VGPR round trip)
// when the gfx1250 builtin is available; plain float4 copy otherwise.
__device__ __forceinline__ void async_copy16(float* lds_dst, const float* gsrc) {
#ifdef HAVE_ASYNC_LDS
    __builtin_amdgcn_global_load_async_to_lds_b128(
        (AS1 vint4*)(uintptr_t)gsrc,
        (AS3 vint4*)(uint32_t)(uintptr_t)lds_dst, 0, 0);
#else
    *(float4*)lds_dst = *(const float4*)gsrc;
#endif
}

__device__ __forceinline__ void wait_async0() {
#ifdef HAVE_ASYNC_LDS
#if __has_builtin(__builtin_amdgcn_s_wait_asynccnt)
    __builtin_amdgcn_s_wait_asynccnt(0);
#else
    asm volatile("s_wait_asynccnt 0" ::: "memory");
#endif
#endif
}

__device__ __forceinline__ float silu_f(float x) {
    return x * __builtin_amdgcn_rcpf(1.0f + __expf(-x));
}

// fp32 atomic max via sign-aware integer atomics
__device__ __forceinline__ void atomicMaxFloat(float* addr, float val) {
    if (val >= 0.0f) atomicMax((int*)addr, __float_as_int(val));
    else             atomicMin((unsigned int*)addr, __float_as_uint(val));
}

// One V_WMMA_F32_16X16X4_F32 step.
// A tile: LDS row-major [16][ldA]   -> lane pair contiguous (ds_load_b64)
// B tile: LDS transposed [N][ldBK]  -> lane pair contiguous (ds_load_b64)
__device__ __forceinline__ v8f wmma4(const float* As, int ldA,
                                     const float* BT, int ldBK,
                                     int kb, int nb, int lane, v8f acc) {
    int row  = lane & 15;           // A row / B column
    int half = lane >> 4;
    v2f a, b;
    a[0] = As[row * ldA + kb + 2 * half + 0];
    a[1] = As[row * ldA + kb + 2 * half + 1];
    b[0] = BT[(nb + row) * ldBK + kb + 2 * half + 0];
    b[1] = BT[(nb + row) * ldBK + kb + 2 * half + 1];
    return __builtin_amdgcn_wmma_f32_16x16x4_f32(false, a, false, b,
                                                 (short)0, acc, false, false);
}

// ---------------- kernel 0: init segment-softmax state ----------------
__global__ void init_kernel(float* mb, float* zb, float* agg) {
    int idx = blockIdx.x * blockDim.x + threadIdx.x;
    if (idx < NN * HEADS_) { mb[idx] = -1e30f; zb[idx] = 0.0f; }
    if (idx < NN * NC_ * 64) agg[idx] = 0.0f;
}

// ---------------- kernel 1: build node features, inv, v_nodes = node @ Wv ----
__global__ void build_node_kernel(const float* __restrict__ bb_emb,
                                  const float* __restrict__ bb_rel,
                                  const unsigned char* __restrict__ mask,
                                  const float* __restrict__ Wv,
                                  float* __restrict__ inv,   // [N][36] (35 + pad0)
                                  float* __restrict__ vno) { // [N][9][64]
    __shared__ float nodeS[NC_ * NODE_C_];    // 9 x 35
    __shared__ float WvS[NODE_C_ * 64];       // 35 x 64
    int n = blockIdx.x, t = threadIdx.x;
    for (int i = t; i < NC_ * NODE_C_; i += 64) {
        int c = i / NODE_C_, k = i - c * NODE_C_;
        float v;
        if (k < 32) {
            v = bb_emb[n * (NC_ * 32) + c * 32 + k];
        } else {
            int b = k - 32;
            if (c == 0)       v = (b == 2) ? (mask[n] ? 1.0f : 0.0f) : 0.0f;
            else if (c <= 3)  v = bb_rel[n * 9 + b * 3 + (c - 1)]; // swapaxes
            else              v = 0.0f;
        }
        nodeS[i] = v;
    }
    for (int i = t; i < NODE_C_ * 64; i += 64) WvS[i] = Wv[i];
    __syncthreads();
    if (t < 36) inv[n * 36 + t] = (t < 35) ? nodeS[t] : 0.0f;
    for (int c = 0; c < NC_; ++c) {
        float acc = 0.0f;
        for (int k = 0; k < NODE_C_; ++k)
            acc += nodeS[c * NODE_C_ + k] * WvS[k * 64 + t];
        vno[n * (NC_ * 64) + c * 64 + t] = acc;
    }
}

// ---------------- kernel 2: attention logits (WMMA) + segment max ------------
// A row layout: [inv36(src) | inv36(dst) | ef64] = 136 cols (pads are zero);
// Wa1 rows remapped to match (zero rows at 35 and 71), stored transposed.
__global__ void attn_logits_kernel(const float* __restrict__ inv,
                                   const float* __restrict__ ef,
                                   const int* __restrict__ ei,
                                   const float* __restrict__ Wa1,  // (134,32)
                                   const float* __restrict__ Wa2,  // (32,8)
                                   float* __restrict__ logits,     // [E][8]
                                   float* __restrict__ mb) {       // [N][8]
    __shared__ alignas(16) float Wa1T[32 * 138];   // [n][k] transposed, k-pad 138
    __shared__ alignas(16) float Wa2T[16 * 34];    // [n][k] transposed, k-pad 34
    __shared__ alignas(16) float AsAll[4][16 * 136];
    int tid = threadIdx.x;
    int wave = tid >> 5, lane = tid & 31;
    float* As = AsAll[wave];
    int e0 = (blockIdx.x * 4 + wave) * 16;
    const int* srcI = ei;
    const int* dstI = ei + EE;

    // issue gathered A tile early (16 edges x 34 16B-chunks, all 16B aligned)
    for (int i = lane; i < 16 * 34; i += 32) {
        int r = i / 34, ch = i - r * 34;
        int e = e0 + r;
        const float* g;
        if (ch < 9)       g = inv + (size_t)srcI[e] * 36 + ch * 4;
        else if (ch < 18) g = inv + (size_t)dstI[e] * 36 + (ch - 9) * 4;
        else              g = ef + (size_t)e * 64 + (ch - 18) * 4;
        async_copy16(&As[r * 136 + ch * 4], g);
    }
    // weights, transposed in LDS; Wa1 rows remapped
    // (0..34 -> 0..34, 35..69 -> 36..70, 70..133 -> 72..135; rows 35,71 zero)
    for (int i = tid; i < 32 * 138; i += blockDim.x) {
        int n2 = i / 138, k = i - n2 * 138;
        float v = 0.0f;
        if (k < 136 && k != 35 && k != 71) {
            int ks = k - (k > 35 ? 1 : 0) - (k > 71 ? 1 : 0);
            v = Wa1[ks * 32 + n2];
        }
        Wa1T[i] = v;
    }
    for (int i = tid; i < 16 * 34; i += blockDim.x) {
        int n2 = i / 34, k = i - n2 * 34;
        Wa2T[i] = (k < 32 && n2 < 8) ? Wa2[k * 8 + n2] : 0.0f;
    }
    wait_async0();
    __syncthreads();

    // GEMM1: 16x136 @ 136x32
    v8f acc0 = {}, acc1 = {};
    for (int kb = 0; kb < 136; kb += 4) {
        acc0 = wmma4(As, 136, Wa1T, 138, kb, 0,  lane, acc0);
        acc1 = wmma4(As, 136, Wa1T, 138, kb, 16, lane, acc1);
    }
    // silu, restage in A-layout (reuse As as [16][32])
    int half = lane >> 4, col = lane & 15;
    for (int r = 0; r < 8; ++r) {
        int m = r + 8 * half;
        As[m * 32 + col]      = silu_f(acc0[r]);
        As[m * 32 + 16 + col] = silu_f(acc1[r]);
    }
    // GEMM2: 16x32 @ 32x16(8 valid)
    v8f acc2 = {};
#pragma unroll
    for (int kb = 0; kb < 32; kb += 4)
        acc2 = wmma4(As, 32, Wa2T, 34, kb, 0, lane, acc2);
    // write logits + segment max
    if (col < 8) {
        for (int r = 0; r < 8; ++r) {
            int m = r + 8 * half;
            int e = e0 + m;
            float val = acc2[r];
            logits[(size_t)e * 8 + col] = val;
            atomicMaxFloat(&mb[dstI[e] * 8 + col], val);
        }
    }
}

// ---------------- kernel 3: w = exp(logit - m[dst]); z[dst] += w --------------
__global__ void softmax_w_kernel(const int* __restrict__ ei,
                                 const float* __restrict__ mb,
                                 float* __restrict__ lw,    // logits in, w out
                                 float* __restrict__ zb) {
    int idx = blockIdx.x * blockDim.x + threadIdx.x;
    if (idx >= EE * HEADS_) return;
    int e = idx >> 3, h = idx & 7;
    int d = ei[EE + e];
    float w = __expf(lw[idx] - mb[d * 8 + h]);
    lw[idx] = w;
    atomicAdd(&zb[d * 8 + h], w);
}

// ---------------- kernel 4: agg[dst] += v_nodes[src] * attn ------------------
__global__ void aggregate_kernel(const int* __restrict__ ei,
                                 const float* __restrict__ w,
                                 const float* __restrict__ zb,
                                 const float* __restrict__ vno,
                                 float* __restrict__ agg) {
    int gw = (blockIdx.x * blockDim.x + threadIdx.x) >> 5;
    int lane = threadIdx.x & 31;
    if (gw >= EE) return;
    int e = gw;
    int s = ei[e], d = ei[EE + e];
    int h0 = lane >> 3, h1 = 4 + (lane >> 3);
    float at0 = w[(size_t)e * 8 + h0] * __builtin_amdgcn_rcpf(zb[d * 8 + h0] + 1e-9f);
    float at1 = w[(size_t)e * 8 + h1] * __builtin_amdgcn_rcpf(zb[d * 8 + h1] + 1e-9f);
    const float* vs = vno + (size_t)s * (NC_ * 64);
    float* ad = agg + (size_t)d * (NC_ * 64);
    for (int c = 0; c < NC_; ++c) {
        atomicAdd(&ad[c * 64 + lane],      vs[c * 64 + lane] * at0);
        atomicAdd(&ad[c * 64 + 32 + lane], vs[c * 64 + 32 + lane] * at1);
    }
}

// ---------------- kernel 5: node update ---------------------------------------
__global__ void node_update_kernel(const float* __restrict__ agg,
                                   const float* __restrict__ Wo,  // (64,32)
                                   const float* __restrict__ W1,  // (32,32)
                                   const float* __restrict__ W2,  // (32,32)
                                   float* __restrict__ out_node,  // [N][9][32]
                                   float* __restrict__ inv2) {    // [N][32]
    __shared__ float WoS[64 * 32], W1S[32 * 32], W2S[32 * 32];
    __shared__ float aggS[NC_ * 64], attnS[NC_ * 32], hS[NC_ * 32], gateS[32];
    int n = blockIdx.x, t = threadIdx.x;
    for (int i = t; i < 64 * 32; i += 32) WoS[i] = Wo[i];
    for (int i = t; i < 32 * 32; i += 32) { W1S[i] = W1[i]; W2S[i] = W2[i]; }
    for (int i = t; i < NC_ * 64; i += 32) aggS[i] = agg[(size_t)n * (NC_ * 64) + i];
    __syncthreads();
    for (int c = 0; c < NC_; ++c) {
        float acc = 0.0f;
        for (int k = 0; k < 64; ++k) acc += aggS[c * 64 + k] * WoS[k * 32 + t];
        attnS[c * 32 + t] = acc;
    }
    __syncthreads();
    for (int c = 0; c < NC_; ++c) {
        float acc = 0.0f;
        for (int k = 0; k < 32; ++k) acc += attnS[c * 32 + k] * W1S[k * 32 + t];
        hS[c * 32 + t] = acc;
    }
    __syncthreads();
    gateS[t] = silu_f(hS[t]);   // gate = silu(h[:,0,:])
    __syncthreads();
    for (int c = 0; c < NC_; ++c) {
        float acc = attnS[c * 32 + t];
        for (int j = 0; j < 32; ++j) acc += hS[c * 32 + j] * gateS[j] * W2S[j * 32 + t];
        out_node[(size_t)n * (NC_ * 32) + c * 32 + t] = acc;
        if (c == 0) inv2[n * 32 + t] = acc;
    }
}

// ---------------- kernel 6: edge output MLP (WMMA) ----------------------------
__global__ void edge_mlp_kernel(const float* __restrict__ inv2,
                                const float* __restrict__ ef,
                                const int* __restrict__ ei,
                                const float* __restrict__ We1,  // (128,32)
                                const float* __restrict__ We2,  // (32,64)
                                float* __restrict__ out_edge) { // [E][64]
    __shared__ alignas(16) float We1T[32 * 130];   // [n][k] transposed, k-pad 130
    __shared__ alignas(16) float We2T[64 * 34];    // [n][k] transposed, k-pad 34
    __shared__ alignas(16) float AsAll[4][16 * 128];
    int tid = threadIdx.x;
    int wave = tid >> 5, lane = tid & 31;
    float* As = AsAll[wave];
    int e0 = (blockIdx.x * 4 + wave) * 16;
    const int* srcI = ei;
    const int* dstI = ei + EE;

    // gathered A tile: 16 edges x 32 16B-chunks, async gather into LDS
    for (int i = lane; i < 16 * 32; i += 32) {
        int r = i >> 5, ch = i & 31;
        int e = e0 + r;
        const float* g;
        if (ch < 8)       g = inv2 + (size_t)srcI[e] * 32 + ch * 4;
        else if (ch < 16) g = inv2 + (size_t)dstI[e] * 32 + (ch - 8) * 4;
        else              g = ef + (size_t)e * 64 + (ch - 16) * 4;
        async_copy16(&As[r * 128 + ch * 4], g);
    }
    // weights, transposed in LDS
    for (int i = tid; i < 32 * 130; i += blockDim.x) {
        int n2 = i / 130, k = i - n2 * 130;
        We1T[i] = (k < 128) ? We1[k * 32 + n2] : 0.0f;
    }
    for (int i = tid; i < 64 * 34; i += blockDim.x) {
        int n2 = i / 34, k = i - n2 * 34;
        We2T[i] = (k < 32) ? We2[k * 64 + n2] : 0.0f;
    }
    wait_async0();
    __syncthreads();

    v8f acc0 = {}, acc1 = {};
    for (int kb = 0; kb < 128; kb += 4) {
        acc0 = wmma4(As, 128, We1T, 130, kb, 0,  lane, acc0);
        acc1 = wmma4(As, 128, We1T, 130, kb, 16, lane, acc1);
    }
    int half = lane >> 4, col = lane & 15;
    for (int r = 0; r < 8; ++r) {
        int m = r + 8 * half;
        As[m * 32 + col]      = silu_f(acc0[r]);
        As[m * 32 + 16 + col] = silu_f(acc1[r]);
    }
    v8f acc[4];
    acc[0] = {}; acc[1] = {}; acc[2] = {}; acc[3] = {};
    for (int kb = 0; kb < 32; kb += 4) {
#pragma unroll
        for (int t2 = 0; t2 < 4; ++t2)
            acc[t2] = wmma4(As, 32, We2T, 34, kb, t2 * 16, lane, acc[t2]);
    }
#pragma unroll
    for (int t2 = 0; t2 < 4; ++t2)
        for (int r = 0; r < 8; ++r) {
            int m = r + 8 * half;
            int e = e0 + m;
            int nn2 = t2 * 16 + col;
            out_edge[(size_t)e * 64 + nn2] = ef[(size_t)e * 64 + nn2] + acc[t2][r];
        }
}

// ---------------- launch ------------------------------------------------------
extern "C" void kernel_launch(void* const* d_in, const int* in_sizes, int n_in,
                              void* d_out, int out_size, void* d_ws, size_t ws_size,
                              hipStream_t stream) {
    (void)in_sizes; (void)n_in; (void)out_size; (void)ws_size;
    const float* bb_emb = (const float*)d_in[0];
    const float* bb_rel = (const float*)d_in[1];
    const float* ef     = (const float*)d_in[2];
    const int*   ei     = (const int*)d_in[3];
    const unsigned char* mask = (const unsigned char*)d_in[4];
    const float* Wa1 = (const float*)d_in[5];
    const float* Wa2 = (const float*)d_in[6];
    const float* Wv  = (const float*)d_in[7];
    const float* Wo  = (const float*)d_in[8];
    const float* W1  = (const float*)d_in[9];
    const float* W2  = (const float*)d_in[10];
    const float* We1 = (const float*)d_in[11];
    const float* We2 = (const float*)d_in[12];

    float* out_node = (float*)d_out;                 // N*9*32
    float* out_edge = out_node + (size_t)NN * 288;   // E*64

    float* ws   = (float*)d_ws;
    float* inv  = ws;                                // N*36
    float* vno  = inv  + (size_t)NN * 36;            // N*576
    float* lw   = vno  + (size_t)NN * 576;           // E*8
    float* mb   = lw   + (size_t)EE * 8;             // N*8
    float* zb   = mb   + (size_t)NN * 8;             // N*8
    float* agg  = zb   + (size_t)NN * 8;             // N*576
    float* inv2 = agg  + (size_t)NN * 576;           // N*32

    init_kernel<<<(NN * 576 + 255) / 256, 256, 0, stream>>>(mb, zb, agg);
    build_node_kernel<<<NN, 64, 0, stream>>>(bb_emb, bb_rel, mask, Wv, inv, vno);
    attn_logits_kernel<<<EE / 64, 128, 0, stream>>>(inv, ef, ei, Wa1, Wa2, lw, mb);
    softmax_w_kernel<<<(EE * 8 + 255) / 256, 256, 0, stream>>>(ei, mb, lw, zb);
    aggregate_kernel<<<EE / 8, 256, 0, stream>>>(ei, lw, zb, vno, agg);
    node_update_kernel<<<NN, 32, 0, stream>>>(agg, Wo, W1, W2, out_node, inv2);
    edge_mlp_kernel<<<EE / 64, 128, 0, stream>>>(inv2, ef, ei, We1, We2, out_edge);
}